// LeakyIntegrateFireNeuron_40518721470625
// MI455X (gfx1250) — compile-verified
//
#include <hip/hip_runtime.h>

// LIF neuron scan: B=32 batches, T=1000 steps, N=1024 neurons.
// Memory-bound streaming kernel (~393 MB total -> ~17us floor at 23.3 TB/s).
// Strategy: one lane owns 4 neuron chains (float4); serial loop over T with a
// 16-deep async global->LDS prefetch pipeline (ASYNCcnt) and non-temporal
// b128 stores for the write-once outputs.

#define BATCH   32
#define NSTEPS  1000
#define NNEUR   1024
#define VEC     4
#define BLOCK   32              // one wave32 per workgroup, no barriers needed
#define PF      16              // prefetch depth (timesteps in flight per wave)
#define NBUF    (PF + 2)        // ring slots; +2 => async write never races ds_load

#if defined(__has_builtin)
#if __has_builtin(__builtin_amdgcn_global_load_async_to_lds_b128) && \
    __has_builtin(__builtin_amdgcn_s_wait_asynccnt)
#define HAVE_ASYNC 1
#endif
#endif
#ifndef HAVE_ASYNC
#define HAVE_ASYNC 0
#endif

// Native vector types (HIP's float4 is a class; builtins need real vectors).
typedef float v4f __attribute__((ext_vector_type(4)));
typedef int   v4i __attribute__((vector_size(4 * sizeof(int))));

typedef __attribute__((address_space(1))) v4i gv4i;   // global int4
typedef __attribute__((address_space(3))) v4i lv4i;   // LDS int4

__device__ __forceinline__ void async_copy16(const float* g, v4f* l) {
#if HAVE_ASYNC
    __builtin_amdgcn_global_load_async_to_lds_b128(
        (gv4i*)g, (lv4i*)l, /*offset=*/0, /*cpol=*/0);
#else
    (void)g; (void)l;
#endif
}

// One Euler step + threshold + reset for 4 packed neurons.
// v' = v + (i - v)/tau ; s = (v' >= 1) ; v'' = s ? 0 : v'
__device__ __forceinline__ void lif_step4(v4f& v, const v4f cur, v4f& s) {
#define STEP1(c)                                           \
    {                                                      \
        float nv = fmaf(cur.c - v.c, 0.1f, v.c);           \
        bool  sp = (nv >= 1.0f);                           \
        s.c = sp ? 1.0f : 0.0f;                            \
        v.c = sp ? 0.0f : nv;                              \
    }
    STEP1(x) STEP1(y) STEP1(z) STEP1(w)
#undef STEP1
}

__global__ __launch_bounds__(BLOCK)
void lif_scan_kernel(const float* __restrict__ in,    // [B, T, N]
                     const float* __restrict__ v0,    // [B, N]
                     float*       __restrict__ out) { // spikes [B,T,N] ++ volts [B,T,N]
    const int lane = threadIdx.x;                        // 0..31
    const int blk  = blockIdx.x;                         // 0..255
    const int chunksPerBatch = NNEUR / (BLOCK * VEC);    // 8
    const int b  = blk / chunksPerBatch;
    const int c  = blk % chunksPerBatch;
    const int n0 = c * (BLOCK * VEC) + lane * VEC;

    const size_t row  = (size_t)NNEUR;
    const size_t base = (size_t)b * NSTEPS * NNEUR + (size_t)n0;
    const float* gin    = in  + base;
    float*       spikes = out + base;
    float*       volts  = out + (size_t)BATCH * NSTEPS * NNEUR + base;

    v4f v = *(const v4f*)(v0 + (size_t)b * NNEUR + n0);

#if HAVE_ASYNC
    __shared__ v4f buf[NBUF][BLOCK];

    // Prologue: launch PF timesteps of async global->LDS copies (512B/wave each).
    #pragma unroll
    for (int t = 0; t < PF; ++t)
        async_copy16(gin + (size_t)t * row, &buf[t][lane]);

    int rd = 0, wr = PF;
    int t = 0;
    for (; t < NSTEPS - PF; ++t) {
        // Oldest outstanding copy (timestep t) must have landed in LDS.
        __builtin_amdgcn_s_wait_asynccnt(PF - 1);
        v4f cur = buf[rd][lane];                          // ds_load_b128
        // Keep the pipe full: fetch timestep t+PF into the next ring slot.
        async_copy16(gin + (size_t)(t + PF) * row, &buf[wr][lane]);

        v4f s;
        lif_step4(v, cur, s);
        __builtin_nontemporal_store(s, (v4f*)(spikes + (size_t)t * row));
        __builtin_nontemporal_store(v, (v4f*)(volts  + (size_t)t * row));

        if (++rd == NBUF) rd = 0;
        if (++wr == NBUF) wr = 0;
    }
    // Drain: everything is in LDS now.
    __builtin_amdgcn_s_wait_asynccnt(0);
    for (; t < NSTEPS; ++t) {
        v4f cur = buf[rd][lane];
        v4f s;
        lif_step4(v, cur, s);
        __builtin_nontemporal_store(s, (v4f*)(spikes + (size_t)t * row));
        __builtin_nontemporal_store(v, (v4f*)(volts  + (size_t)t * row));
        if (++rd == NBUF) rd = 0;
    }
#else
    // Fallback: direct streaming loads; unroll so the compiler batches b128 loads.
    #pragma unroll 4
    for (int t = 0; t < NSTEPS; ++t) {
        v4f cur = __builtin_nontemporal_load((const v4f*)(gin + (size_t)t * row));
        v4f s;
        lif_step4(v, cur, s);
        __builtin_nontemporal_store(s, (v4f*)(spikes + (size_t)t * row));
        __builtin_nontemporal_store(v, (v4f*)(volts  + (size_t)t * row));
    }
#endif
}

extern "C" void kernel_launch(void* const* d_in, const int* in_sizes, int n_in,
                              void* d_out, int out_size, void* d_ws, size_t ws_size,
                              hipStream_t stream) {
    (void)in_sizes; (void)n_in; (void)out_size; (void)d_ws; (void)ws_size;
    const float* input  = (const float*)d_in[0];   // [32, 1000, 1024] f32
    const float* v_init = (const float*)d_in[1];   // [32, 1024] f32
    float*       out    = (float*)d_out;           // spikes ++ voltages

    const int grid = BATCH * (NNEUR / (BLOCK * VEC));  // 32 * 8 = 256 blocks
    lif_scan_kernel<<<grid, BLOCK, 0, stream>>>(input, v_init, out);
}